// ResidualGcn_37340445672257
// MI455X (gfx1250) — compile-verified
//
#include <hip/hip_runtime.h>
#include <hip/hip_bf16.h>

#define BB 2
#define NN 1024
#define FF 32
#define CC 32
#define EE 65
#define EPS 1e-3f

typedef __attribute__((ext_vector_type(2))) float v2f;
typedef __attribute__((ext_vector_type(8))) float v8f;

// ---------------- Kernel 1: x = relu(bn(nodes)); xwT[b][c][n] = sum_f x*W ----------
__global__ void k_xwT(const float* __restrict__ nodes,
                      const float* __restrict__ g, const float* __restrict__ be,
                      const float* __restrict__ mu, const float* __restrict__ var,
                      const float* __restrict__ W, float* __restrict__ xwT) {
    int idx = blockIdx.x * blockDim.x + threadIdx.x;   // over B*C*N
    if (idx >= BB * CC * NN) return;
    int n = idx & (NN - 1);
    int c = (idx >> 10) & (CC - 1);
    int b = idx >> 15;
    const float* nrow = nodes + ((size_t)(b * NN + n)) * FF;
    float acc = 0.f;
#pragma unroll
    for (int f = 0; f < FF; ++f) {
        float s = g[f] * rsqrtf(var[f] + EPS);
        float x = fmaf(nrow[f], s, be[f] - mu[f] * s);
        x = fmaxf(x, 0.f);
        acc = fmaf(x, W[f * CC + c], acc);
    }
    xwT[((size_t)(b * CC + c)) * NN + n] = acc;
}

// ---------------- Kernel 2: dinv[b][i] = rsqrt(1 + sum_j clip(adj,0,1)) -----------
__global__ void k_dinv(const float* __restrict__ adj, float* __restrict__ dinv) {
    int row = blockIdx.x;                       // b*N + i
    const float* arow = adj + (size_t)row * NN;
    float s = 0.f;
    for (int j = threadIdx.x; j < NN; j += 256)
        s += fminf(fmaxf(arow[j], 0.f), 1.f);
#pragma unroll
    for (int off = 16; off > 0; off >>= 1)
        s += __shfl_down(s, off, 32);
    __shared__ float red[8];
    if ((threadIdx.x & 31) == 0) red[threadIdx.x >> 5] = s;
    __syncthreads();
    if (threadIdx.x == 0) {
        float d = 1.0f;                          // +I diagonal contribution
#pragma unroll
        for (int w = 0; w < 8; ++w) d += red[w];
        dinv[row] = (d > 0.f) ? rsqrtf(d) : 0.f;
    }
}

// ---------------- Kernel 3: new_nodes = A_hat @ xw + b ; out_nodes = nodes + new --
// One wave per 16-row M tile; f32 WMMA 16x16x4, K-loop over N, two 16-col N tiles.
// The +I diagonal term is applied in closed form in the epilogue:
//   new_nodes[i][c] += dinv[i]^2 * xw[i][c]
// so the hot loop is pure clamp -> scale -> WMMA (no per-element compares).
__global__ __launch_bounds__(256) void k_gcn_wmma(
        const float* __restrict__ adj, const float* __restrict__ xwT,
        const float* __restrict__ dinv, const float* __restrict__ gcn_b,
        const float* __restrict__ nodes,
        float* __restrict__ new_nodes, float* __restrict__ out_nodes) {
    int wave = (blockIdx.x * blockDim.x + threadIdx.x) >> 5;  // 0 .. B*64-1
    int lane = threadIdx.x & 31;
    int b  = wave >> 6;
    int i0 = (wave & 63) * 16;
    int r    = lane & 15;        // A: row within tile ; B: column within tile
    int half = lane >> 4;        // selects K pair {0,1} vs {2,3}
    int row  = i0 + r;           // global row index i for this lane's A elements
    const float* arow = adj + ((size_t)(b * NN + row)) * NN;
    const float  di   = dinv[b * NN + row];
    const float* dj   = dinv + b * NN;
    const float* bt0  = xwT + ((size_t)(b * CC + r)) * NN;        // c = r
    const float* bt1  = xwT + ((size_t)(b * CC + r + 16)) * NN;   // c = r+16

    v8f acc0 = {}; v8f acc1 = {};
    for (int k = 0; k < NN; k += 4) {
        int ka = k + half * 2;
        // A fragment: dinv_i * clip(adj[row][ka..ka+1]) * dinv_j  (no diag branch)
        float a0 = fminf(fmaxf(arow[ka],     0.f), 1.f);
        float a1 = fminf(fmaxf(arow[ka + 1], 0.f), 1.f);
        v2f A; A.x = di * dj[ka] * a0;  A.y = di * dj[ka + 1] * a1;
        // B fragments: xw[ka][c], xw[ka+1][c]  (contiguous in xwT)
        v2f B0; B0.x = bt0[ka]; B0.y = bt0[ka + 1];
        v2f B1; B1.x = bt1[ka]; B1.y = bt1[ka + 1];
        acc0 = __builtin_amdgcn_wmma_f32_16x16x4_f32(false, A, false, B0,
                                                     (short)0, acc0, false, false);
        acc1 = __builtin_amdgcn_wmma_f32_16x16x4_f32(false, A, false, B1,
                                                     (short)0, acc1, false, false);
    }
    // Epilogue: D layout — VGPR v: lanes0-15 -> M=v, lanes16-31 -> M=v+8; N=lane&15
    int col = lane & 15;
    int roff = 8 * half;
    float gb0 = gcn_b[col], gb1 = gcn_b[col + 16];
    const float* xc0 = xwT + ((size_t)(b * CC + col)) * NN;       // xw[.][col]
    const float* xc1 = xwT + ((size_t)(b * CC + col + 16)) * NN;  // xw[.][col+16]
#pragma unroll
    for (int v = 0; v < 8; ++v) {
        int rr = i0 + v + roff;
        float drr = dinv[b * NN + rr];
        float dd  = drr * drr;                    // identity (+I) correction
        size_t o = ((size_t)(b * NN + rr)) * CC + col;
        float v0 = acc0[v] + gb0 + dd * xc0[rr];
        float v1 = acc1[v] + gb1 + dd * xc1[rr];
        new_nodes[o]      = v0;
        new_nodes[o + 16] = v1;
        out_nodes[o]      = nodes[o]      + v0;
        out_nodes[o + 16] = nodes[o + 16] + v1;
    }
}

// ---------------- Kernel 4: Si/Sj per-node sums of BN->ReLU->conv channels --------
__global__ void k_nodesums(const float* __restrict__ nn,
                           const float* __restrict__ eg, const float* __restrict__ eb,
                           const float* __restrict__ em, const float* __restrict__ ev,
                           const float* __restrict__ ecw,
                           float* __restrict__ Si, float* __restrict__ Sj) {
    int idx = blockIdx.x * blockDim.x + threadIdx.x;   // over B*N
    if (idx >= BB * NN) return;
    const float* nrow = nn + (size_t)idx * CC;
    float si = 0.f, sj = 0.f;
#pragma unroll 4
    for (int c = 0; c < CC; ++c) {
        float v = nrow[c];
        int k1 = 1 + c, k2 = 1 + CC + c;
        float s1 = eg[k1] * rsqrtf(ev[k1] + EPS);
        float s2 = eg[k2] * rsqrtf(ev[k2] + EPS);
        si = fmaf(fmaxf(fmaf(v, s1, eb[k1] - em[k1] * s1), 0.f), ecw[k1], si);
        sj = fmaf(fmaxf(fmaf(v, s2, eb[k2] - em[k2] * s2), 0.f), ecw[k2], sj);
    }
    Si[idx] = si; Sj[idx] = sj;
}

// ---------------- Kernel 5: out_adj = adj + relu(s0*adj+t0)*w0 + cb + Si_i + Sj_j -
__global__ void k_edges(const float* __restrict__ adj,
                        const float* __restrict__ Si, const float* __restrict__ Sj,
                        const float* __restrict__ eg, const float* __restrict__ eb,
                        const float* __restrict__ em, const float* __restrict__ ev,
                        const float* __restrict__ ecw, const float* __restrict__ ecb,
                        float* __restrict__ out_adj) {
    size_t q = (size_t)blockIdx.x * blockDim.x + threadIdx.x;  // over B*N*N/4
    if (q >= (size_t)BB * NN * NN / 4) return;
    size_t base = q * 4;
    int b = (int)(base >> 20);
    size_t rem = base & ((size_t)NN * NN - 1);
    int i = (int)(rem >> 10);
    int j = (int)(rem & (NN - 1));
    float s0 = eg[0] * rsqrtf(ev[0] + EPS);
    float t0 = eb[0] - em[0] * s0;
    float w0 = ecw[0];
    float cb = ecb[0];
    float4 a = *(const float4*)(adj + base);
    float si = Si[b * NN + i];
    const float* sjp = Sj + b * NN + j;
    float4 o;
    o.x = a.x + fmaf(fmaxf(fmaf(a.x, s0, t0), 0.f), w0, cb + si + sjp[0]);
    o.y = a.y + fmaf(fmaxf(fmaf(a.y, s0, t0), 0.f), w0, cb + si + sjp[1]);
    o.z = a.z + fmaf(fmaxf(fmaf(a.z, s0, t0), 0.f), w0, cb + si + sjp[2]);
    o.w = a.w + fmaf(fmaxf(fmaf(a.w, s0, t0), 0.f), w0, cb + si + sjp[3]);
    *(float4*)(out_adj + base) = o;
}

extern "C" void kernel_launch(void* const* d_in, const int* in_sizes, int n_in,
                              void* d_out, int out_size, void* d_ws, size_t ws_size,
                              hipStream_t stream) {
    const float* nodes    = (const float*)d_in[0];
    const float* adj      = (const float*)d_in[1];
    const float* bn_gamma = (const float*)d_in[2];
    const float* bn_beta  = (const float*)d_in[3];
    const float* bn_mean  = (const float*)d_in[4];
    const float* bn_var   = (const float*)d_in[5];
    const float* gcn_w    = (const float*)d_in[6];
    const float* gcn_b    = (const float*)d_in[7];
    const float* ebn_g    = (const float*)d_in[8];
    const float* ebn_b    = (const float*)d_in[9];
    const float* ebn_m    = (const float*)d_in[10];
    const float* ebn_v    = (const float*)d_in[11];
    const float* ec_w     = (const float*)d_in[12];
    const float* ec_b     = (const float*)d_in[13];

    float* out_nodes = (float*)d_out;                        // [B,N,F]
    float* out_adj   = (float*)d_out + (size_t)BB * NN * FF; // [B,N,N]

    // Scratch layout (floats)
    float* ws   = (float*)d_ws;
    float* xwT  = ws;                                  // B*C*N   = 65536
    float* dinv = xwT + (size_t)BB * CC * NN;          // B*N     =  2048
    float* nn   = dinv + (size_t)BB * NN;              // B*N*C   = 65536
    float* Si   = nn + (size_t)BB * NN * CC;           // B*N     =  2048
    float* Sj   = Si + (size_t)BB * NN;                // B*N     =  2048

    // 1) XW (transposed for WMMA B-operand contiguity)
    k_xwT<<<(BB * CC * NN + 255) / 256, 256, 0, stream>>>(
        nodes, bn_gamma, bn_beta, bn_mean, bn_var, gcn_w, xwT);
    // 2) degrees
    k_dinv<<<BB * NN, 256, 0, stream>>>(adj, dinv);
    // 3) WMMA GEMM + node residual (B*64 waves, 8 waves/block)
    k_gcn_wmma<<<(BB * (NN / 16) * 32) / 256, 256, 0, stream>>>(
        adj, xwT, dinv, gcn_b, nodes, nn, out_nodes);
    // 4) per-node edge-channel sums
    k_nodesums<<<(BB * NN + 255) / 256, 256, 0, stream>>>(
        nn, ebn_g, ebn_b, ebn_m, ebn_v, ec_w, Si, Sj);
    // 5) edge residual pass (float4 vectorized)
    k_edges<<<(int)(((size_t)BB * NN * NN / 4 + 255) / 256), 256, 0, stream>>>(
        adj, Si, Sj, ebn_g, ebn_b, ebn_m, ebn_v, ec_w, ec_b, out_adj);
}